// PoseEncoding_14044543058049
// MI455X (gfx1250) — compile-verified
//
#include <hip/hip_runtime.h>

// ---------------------------------------------------------------------------
// out[b,s,n,j] = x[b,s,n,j] + pe[n,j]   (B=8,S=16,N=1024,d=256, f32)
// Bandwidth-bound: 256 MiB stream @ 23.3 TB/s ~ 11.5us. Strategy:
//   * precompute 1 MiB pe table once (d_ws)
//   * per block: async-copy a 16 KiB pe tile global->LDS (CDNA5 async path),
//     reuse it across 8 broadcast (b,s) slices
//   * stream x/out with non-temporal float4 accesses (don't pollute L2)
// ---------------------------------------------------------------------------

typedef float v4f __attribute__((ext_vector_type(4)));

#define B_  8
#define S_  16
#define N_  1024
#define D_  256
#define ND_ (N_ * D_)          // 262144 floats, 1 MiB
#define BS_ (B_ * S_)          // 128 broadcast slices

#define THREADS        256
#define TILE4          1024                 // float4 per tile -> 4096 floats, 16 KiB LDS
#define TILE_FLOATS    (TILE4 * 4)
#define NTILES         (ND_ / TILE_FLOATS)  // 64
#define BS_PER_BLOCK   8
#define NBSG           (BS_ / BS_PER_BLOCK) // 16
#define V4_PER_THREAD  (TILE4 / THREADS)    // 4

__device__ __forceinline__ float pe_value(int flat_idx) {
    int n = flat_idx >> 8;          // / D_ (D_ == 256)
    int j = flat_idx & (D_ - 1);
    int k = j >> 1;
    float denom = 1000.0f * (2.0f * ((float)k / (float)D_)) + 0.01f;
    float ang = (float)n / denom;
    return (j & 1) ? __builtin_cosf(ang) : __builtin_sinf(ang);
}

// Kernel 1: build the 1 MiB pe table in workspace (amortized over everything).
__global__ void __launch_bounds__(THREADS)
pe_table_kernel(float* __restrict__ pe) {
    int i = blockIdx.x * THREADS + threadIdx.x;
    if (i < ND_) pe[i] = pe_value(i);
}

// Kernel 2: stream x + pe -> out.
__global__ void __launch_bounds__(THREADS)
add_pe_kernel(const float* __restrict__ x,
              const float* __restrict__ pe,
              float* __restrict__ out,
              int use_ws) {
    __shared__ v4f tile[TILE4];     // 16 KiB of the pe table, block-resident

    const int t        = threadIdx.x;
    const int tileIdx  = blockIdx.x;            // 0..NTILES-1
    const int bsg      = blockIdx.y;            // 0..NBSG-1
    const long tileBase = (long)tileIdx * TILE_FLOATS;   // float offset into pe

    if (use_ws) {
        // CDNA5 async copy: per-lane global_load_async_to_lds_b128.
        // LDS dest addr = low 32 bits of the generic LDS pointer (wave-relative).
        const float* peBase = pe + tileBase;
#pragma unroll
        for (int u = 0; u < V4_PER_THREAD; ++u) {
            int idx4 = t + u * THREADS;
            unsigned ldsAddr = (unsigned)(unsigned long long)(&tile[idx4]);
            unsigned voff    = (unsigned)(idx4 * 16);
            asm volatile("global_load_async_to_lds_b128 %0, %1, %2"
                         :: "v"(ldsAddr), "v"(voff), "s"(peBase)
                         : "memory");
        }
        asm volatile("s_wait_asynccnt 0" ::: "memory");
    } else {
        // Fallback (tiny ws): synthesize the tile with v_sin/v_cos, once per
        // block, amortized over BS_PER_BLOCK slices.
#pragma unroll
        for (int u = 0; u < V4_PER_THREAD; ++u) {
            int idx4 = t + u * THREADS;
            v4f v;
#pragma unroll
            for (int e = 0; e < 4; ++e)
                v[e] = pe_value((int)tileBase + idx4 * 4 + e);
            tile[idx4] = v;
        }
    }
    __syncthreads();

    const int bs0 = bsg * BS_PER_BLOCK;
#pragma unroll 1
    for (int r = 0; r < BS_PER_BLOCK; ++r) {
        const long base = (long)(bs0 + r) * ND_ + tileBase;  // floats
        const v4f* __restrict__ xp = (const v4f*)(x + base);
        v4f*       __restrict__ op = (v4f*)(out + base);
#pragma unroll
        for (int u = 0; u < V4_PER_THREAD; ++u) {
            int idx4 = t + u * THREADS;
            v4f xv = __builtin_nontemporal_load(&xp[idx4]);  // streamed once
            v4f pv = tile[idx4];                             // LDS-resident
            v4f s  = xv + pv;
            __builtin_nontemporal_store(s, &op[idx4]);       // streamed once
        }
    }
}

extern "C" void kernel_launch(void* const* d_in, const int* in_sizes, int n_in,
                              void* d_out, int out_size, void* d_ws, size_t ws_size,
                              hipStream_t stream) {
    const float* x   = (const float*)d_in[0];
    float*       out = (float*)d_out;
    float*       pe  = (float*)d_ws;

    const int use_ws = (ws_size >= (size_t)ND_ * sizeof(float)) ? 1 : 0;

    if (use_ws) {
        pe_table_kernel<<<ND_ / THREADS, THREADS, 0, stream>>>(pe);
    }
    dim3 grid(NTILES, NBSG);
    add_pe_kernel<<<grid, THREADS, 0, stream>>>(x, pe, out, use_ws);
}